// AntColonyOptimizer_50964081934651
// MI455X (gfx1250) — compile-verified
//
#include <hip/hip_runtime.h>
#include <stdint.h>

#define H_DIM 2048
#define A_ANTS 8192
#define B_DIM 8
#define S_DIM 2048

typedef __attribute__((ext_vector_type(2))) float v2f;
typedef __attribute__((ext_vector_type(8))) float v8f;

__device__ __forceinline__ uint32_t rotl32(uint32_t x, int r) {
  return (x << r) | (x >> (32 - r));
}

// Exact JAX threefry2x32 (20 rounds), key = (0, 42) for jax.random.key(42)
__device__ __forceinline__ void threefry2x32(uint32_t k0, uint32_t k1,
                                             uint32_t c0, uint32_t c1,
                                             uint32_t& o0, uint32_t& o1) {
  uint32_t ks0 = k0, ks1 = k1, ks2 = k0 ^ k1 ^ 0x1BD11BDAu;
  uint32_t x0 = c0 + ks0, x1 = c1 + ks1;
  const int rot0[4] = {13, 15, 26, 6};
  const int rot1[4] = {17, 29, 16, 24};
#pragma unroll
  for (int i = 0; i < 5; ++i) {
    const int* rr = (i & 1) ? rot1 : rot0;
#pragma unroll
    for (int j = 0; j < 4; ++j) {
      x0 += x1; x1 = rotl32(x1, rr[j]); x1 ^= x0;
    }
    uint32_t ka = (i % 3 == 0) ? ks1 : (i % 3 == 1) ? ks2 : ks0; // ks[(i+1)%3]
    uint32_t kb = (i % 3 == 0) ? ks2 : (i % 3 == 1) ? ks0 : ks1; // ks[(i+2)%3]
    x0 += ka; x1 += kb + (uint32_t)(i + 1);
  }
  o0 = x0; o1 = x1;
}

// Gumbel noise for flat element i of the [A,H] draw, matching JAX's
// split-iota counter layout: bits[i] (i<n/2) = out0 of ctr (i, i+n/2), else out1.
__device__ __forceinline__ float gumbel_at(uint32_t i) {
  const uint32_t n2 = (uint32_t)A_ANTS * (uint32_t)H_DIM / 2u; // 8388608
  bool second = i >= n2;
  uint32_t j = second ? (i - n2) : i;
  uint32_t o0, o1;
  threefry2x32(0u, 42u, j, j + n2, o0, o1);
  uint32_t bits = second ? o1 : o0;
  uint32_t fb = (bits >> 9) | 0x3f800000u;
  float u01 = __uint_as_float(fb) - 1.0f;            // [0,1)
  const float TINY = 1.1754944e-38f;
  float u = u01 * (1.0f - TINY) + TINY;              // uniform(minval=tiny)
  u = fmaxf(u, TINY);
  return -__logf(-__logf(u));
}

// ---------------- kernels ----------------

__global__ void k_init(float* __restrict__ diag) {
  int h = blockIdx.x * blockDim.x + threadIdx.x;
  if (h < H_DIM) diag[h] = 0.0f;
}

// One wave per ant: Gumbel-max categorical over trails[pos[a], :]
__global__ void k_sample(const float* __restrict__ trails,
                         const int* __restrict__ ant_pos,
                         int* __restrict__ next_pos_out) {
  int wave = (blockIdx.x * blockDim.x + threadIdx.x) >> 5;
  int lane = threadIdx.x & 31;
  if (wave >= A_ANTS) return;
  int pos = ant_pos[wave];
  const float* row = trails + (size_t)pos * H_DIM;
  float bv = -3.4e38f; int bh = 0;
  for (int h = lane; h < H_DIM; h += 32) {
    float v = row[h] + gumbel_at((uint32_t)wave * (uint32_t)H_DIM + (uint32_t)h);
    if (v > bv) { bv = v; bh = h; }   // ascending h -> first-max tie-break in-lane
  }
  for (int off = 16; off > 0; off >>= 1) {
    float ov = __shfl_down(bv, off, 32);
    int   oh = __shfl_down(bh, off, 32);
    if (ov > bv || (ov == bv && oh < bh)) { bv = ov; bh = oh; }
  }
  if (lane == 0) next_pos_out[wave] = bh;
}

// One block per ant row: new_paths copy + scatter(1.0) + row L2 norm, fused
__global__ void k_paths(const float* __restrict__ ant_paths,
                        const int* __restrict__ next_pos,
                        float* __restrict__ new_paths,
                        float* __restrict__ len,
                        float* __restrict__ upd,
                        const float* __restrict__ strength) {
  int a = blockIdx.x;
  int t = threadIdx.x;
  int npos = next_pos[a];
  const float4* src = (const float4*)(ant_paths + (size_t)a * H_DIM);
  float4* dst = (float4*)(new_paths + (size_t)a * H_DIM);
  float acc = 0.0f;
#pragma unroll
  for (int j = 0; j < 2; ++j) {
    int i4 = t + 256 * j;
    float4 v = src[i4];
    int hb = i4 * 4;
    if (npos >= hb && npos < hb + 4) ((float*)&v)[npos - hb] = 1.0f;
    acc += v.x * v.x + v.y * v.y + v.z * v.z + v.w * v.w;
    dst[i4] = v;
  }
  __shared__ float s[256];
  s[t] = acc;
  __syncthreads();
  for (int o = 128; o > 0; o >>= 1) {
    if (t < o) s[t] += s[t + o];
    __syncthreads();
  }
  if (t == 0) {
    float l = sqrtf(s[0]);
    len[a] = l;
    upd[a] = strength[0] / (l + 1e-8f);
  }
}

// First-occurrence argmin over path lengths, single block
__global__ void k_best(const float* __restrict__ len,
                       const float* __restrict__ best_path_length,
                       float* __restrict__ out_bestlen,
                       int* __restrict__ meta) {
  __shared__ float sv[256];
  __shared__ int   si[256];
  int t = threadIdx.x;
  float mv = 3.4e38f; int mi = 0x7fffffff;
  for (int i = t; i < A_ANTS; i += 256) {
    float v = len[i];
    if (v < mv) { mv = v; mi = i; }
  }
  sv[t] = mv; si[t] = mi;
  __syncthreads();
  for (int o = 128; o > 0; o >>= 1) {
    if (t < o) {
      if (sv[t + o] < sv[t] || (sv[t + o] == sv[t] && si[t + o] < si[t])) {
        sv[t] = sv[t + o]; si[t] = si[t + o];
      }
    }
    __syncthreads();
  }
  if (t == 0) {
    float m = sv[0]; int a = si[0];
    float b = best_path_length[0];
    int improved = (m < b) ? 1 : 0;
    out_bestlen[0] = improved ? m : b;
    meta[0] = a;
    meta[1] = improved;
  }
}

__global__ void k_bestpath(const float* __restrict__ new_paths,
                           const float* __restrict__ best_path,
                           const int* __restrict__ meta,
                           float* __restrict__ bestvec) {
  int h = blockIdx.x * blockDim.x + threadIdx.x;
  if (h >= H_DIM) return;
  int a = meta[0], improved = meta[1];
  bestvec[h] = improved ? new_paths[(size_t)a * H_DIM + h] : best_path[h];
}

// diag_add[h] = sum_a update[a] * (new_paths[a,h] > 0)
// GEMV as WMMA f32 16x16x4: A = update broadcast across 16 rows, B = mask tile.
// A layout (16x4 f32): v0 = K{0 | 2}, v1 = K{1 | 3} (lanes 0-15 | 16-31).
__global__ void k_diag(const float* __restrict__ new_paths,
                       const float* __restrict__ upd,
                       float* __restrict__ diag) {
  int gw   = (blockIdx.x * blockDim.x + threadIdx.x) >> 5; // 0..2047
  int lane = threadIdx.x & 31;
  int col_tile = gw & 127;   // 128 tiles of 16 columns
  int part     = gw >> 7;    // 16 row splits
  int h0   = col_tile * 16;
  int half = lane >> 4;
  int l16  = lane & 15;
  int aBeg = part * (A_ANTS / 16);
  int aEnd = aBeg + (A_ANTS / 16);
#if __has_builtin(__builtin_amdgcn_wmma_f32_16x16x4_f32)
  v8f c = {};
  for (int a0 = aBeg; a0 < aEnd; a0 += 4) {
    int ra = a0 + 2 * half;
    v2f av; av.x = upd[ra]; av.y = upd[ra + 1];
    float m0 = (new_paths[(size_t)ra       * H_DIM + h0 + l16] > 0.0f) ? 1.0f : 0.0f;
    float m1 = (new_paths[(size_t)(ra + 1) * H_DIM + h0 + l16] > 0.0f) ? 1.0f : 0.0f;
    v2f bv; bv.x = m0; bv.y = m1;
    c = __builtin_amdgcn_wmma_f32_16x16x4_f32(false, av, false, bv,
                                              (short)0, c, false, false);
  }
  // C/D layout: VGPR0, lanes 0-15 -> M=0, N=lane (all D rows identical here)
  if (lane < 16) atomicAdd(&diag[h0 + lane], c[0]);
#else
  float acc = 0.0f;
  for (int a = aBeg; a < aEnd; ++a) {
    float m = new_paths[(size_t)a * H_DIM + h0 + l16];
    acc += (m > 0.0f) ? upd[a] : 0.0f;
  }
  if (half == 0) atomicAdd(&diag[h0 + l16], acc);
#endif
}

// new_trails = (trails + diag-scatter) * (1 - decay), float4 streamed
__global__ void k_trails(const float* __restrict__ trails,
                         const float* __restrict__ diag,
                         const float* __restrict__ decay,
                         float* __restrict__ out) {
  int i4 = blockIdx.x * blockDim.x + threadIdx.x; // H*H/4 = 1048576 total
  const float4* src = (const float4*)trails;
  float4* dst = (float4*)out;
  float4 v = src[i4];
  int row  = i4 >> 9;            // H/4 = 512 float4 per row
  int col0 = (i4 & 511) * 4;
  float d  = 1.0f - decay[0];
  float add = diag[row];
  float* vp = (float*)&v;
#pragma unroll
  for (int c = 0; c < 4; ++c) {
    float val = vp[c] + ((col0 + c == row) ? add : 0.0f);
    vp[c] = val * d;
  }
  dst[i4] = v;
}

// output = x * best_path (broadcast over B,S); dominant 256 MiB stream
__global__ void k_scale(const float* __restrict__ x,
                        const float* __restrict__ bestvec,
                        float* __restrict__ out) {
  const float4* x4 = (const float4*)x;
  const float4* b4 = (const float4*)bestvec;
  float4* o4 = (float4*)out;
  const int n4 = (B_DIM * S_DIM * H_DIM) / 4; // 8388608
  int stride = gridDim.x * blockDim.x;
  for (int i = blockIdx.x * blockDim.x + threadIdx.x; i < n4; i += stride) {
    __builtin_prefetch(&x4[i + stride], 0, 1); // global_prefetch_b8
    float4 xv = x4[i];
    float4 bv = b4[i & 511];                   // H/4 = 512
    xv.x *= bv.x; xv.y *= bv.y; xv.z *= bv.z; xv.w *= bv.w;
    o4[i] = xv;
  }
}

// ---------------- launch ----------------

extern "C" void kernel_launch(void* const* d_in, const int* in_sizes, int n_in,
                              void* d_out, int out_size, void* d_ws, size_t ws_size,
                              hipStream_t stream) {
  (void)in_sizes; (void)n_in; (void)out_size; (void)ws_size;
  const float* x        = (const float*)d_in[0];
  const float* trails   = (const float*)d_in[1];
  const float* paths    = (const float*)d_in[2];
  const float* bestp    = (const float*)d_in[3];
  const float* bpl      = (const float*)d_in[4];
  const float* decay    = (const float*)d_in[5];
  const float* strength = (const float*)d_in[6];
  const int*   antpos   = (const int*)d_in[7];

  float* out        = (float*)d_out;
  float* out_x      = out;                                  // 33554432 f32
  float* out_trails = out + 33554432;                       //  4194304 f32
  float* out_paths  = out + 33554432 + 4194304;             // 16777216 f32
  float* out_blen   = out + 54525952;                       //        1 f32
  int*   out_next   = (int*)(out + 54525953);               //     8192 i32

  float* ws      = (float*)d_ws;
  float* ws_len  = ws;                // [8192]
  float* ws_upd  = ws + 8192;         // [8192]
  float* ws_diag = ws + 16384;        // [2048]
  float* ws_best = ws + 18432;        // [2048]
  int*   ws_meta = (int*)(ws + 20480);// [2]

  k_init    <<<8,    256, 0, stream>>>(ws_diag);
  k_sample  <<<1024, 256, 0, stream>>>(trails, antpos, out_next);
  k_paths   <<<8192, 256, 0, stream>>>(paths, out_next, out_paths, ws_len, ws_upd, strength);
  k_best    <<<1,    256, 0, stream>>>(ws_len, bpl, out_blen, ws_meta);
  k_bestpath<<<8,    256, 0, stream>>>(out_paths, bestp, ws_meta, ws_best);
  k_diag    <<<256,  256, 0, stream>>>(out_paths, ws_upd, ws_diag);
  k_trails  <<<4096, 256, 0, stream>>>(trails, ws_diag, decay, out_trails);
  k_scale   <<<8192, 256, 0, stream>>>(x, ws_best, out_x);
}